// TransitionDown_1400159339077
// MI455X (gfx1250) — compile-verified
//
#include <hip/hip_runtime.h>
#include <hip/hip_bf16.h>

#define B_    8
#define N_    8192
#define CIN   64
#define COUT  128
#define M_    2048
#define KNN_  16
#define EPS_  1e-5f

typedef __attribute__((ext_vector_type(2))) float v2f;
typedef __attribute__((ext_vector_type(8))) float v8f;

// ---------------------------------------------------------------------------
// 1) Farthest point sampling: one 1024-thread workgroup per batch.
//    Cloud (x,y,z) + running min-dist kept fully in LDS (128KB < 320KB/WGP).
//    Writes p_out (B,M,3) directly into d_out tail.
// ---------------------------------------------------------------------------
__global__ __launch_bounds__(1024) void fps_kernel(const float* __restrict__ p,
                                                   float* __restrict__ p_out)
{
    extern __shared__ float smem[];
    float* px    = smem;                       // N_
    float* py    = px + N_;                    // N_
    float* pz    = py + N_;                    // N_
    float* dist  = pz + N_;                    // N_
    float* red_d = dist + N_;                  // 1024
    int*   red_i = (int*)(red_d + 1024);       // 1024
    float* bcast = (float*)(red_i + 1024);     // 3

    const int b   = blockIdx.x;
    const int tid = threadIdx.x;
    const float* pb = p + (size_t)b * N_ * 3;

    for (int s = 0; s < N_ / 1024; ++s) {
        int i = tid + s * 1024;
        px[i] = pb[i * 3 + 0];
        py[i] = pb[i * 3 + 1];
        pz[i] = pb[i * 3 + 2];
        dist[i] = __builtin_inff();
    }
    __syncthreads();

    if (tid == 0) {
        bcast[0] = px[0]; bcast[1] = py[0]; bcast[2] = pz[0];
        float* po = p_out + (size_t)b * M_ * 3;
        po[0] = px[0]; po[1] = py[0]; po[2] = pz[0];
    }
    __syncthreads();

    for (int j = 1; j < M_; ++j) {
        const float lx = bcast[0], ly = bcast[1], lz = bcast[2];
        float bestv = -1.0f; int besti = 0x7fffffff;
        for (int s = 0; s < N_ / 1024; ++s) {
            int i = tid + s * 1024;
            float dx = px[i] - lx, dy = py[i] - ly, dz = pz[i] - lz;
            float d = __fadd_rn(__fadd_rn(__fmul_rn(dx, dx), __fmul_rn(dy, dy)),
                                __fmul_rn(dz, dz));
            float nd = fminf(dist[i], d);
            dist[i] = nd;
            if (nd > bestv || (nd == bestv && i < besti)) { bestv = nd; besti = i; }
        }
        red_d[tid] = bestv; red_i[tid] = besti;
        __syncthreads();
        if (tid < 32) {
            float v = red_d[tid]; int ii = red_i[tid];
            for (int q = 1; q < 32; ++q) {
                float ov = red_d[tid + q * 32]; int oi = red_i[tid + q * 32];
                if (ov > v || (ov == v && oi < ii)) { v = ov; ii = oi; }
            }
            for (int off = 16; off > 0; off >>= 1) {
                float ov = __shfl_xor(v, off, 32);
                int   oi = __shfl_xor(ii, off, 32);
                if (ov > v || (ov == v && oi < ii)) { v = ov; ii = oi; }
            }
            if (tid == 0) {
                float sx = px[ii], sy = py[ii], sz = pz[ii];
                bcast[0] = sx; bcast[1] = sy; bcast[2] = sz;
                float* po = p_out + ((size_t)b * M_ + j) * 3;
                po[0] = sx; po[1] = sy; po[2] = sz;
            }
        }
        __syncthreads();
    }
}

// ---------------------------------------------------------------------------
// 2) Pointwise MLP via V_WMMA_F32_16X16X4_F32 + fused BN(eval) + ReLU.
//    (B*N, 64) x (64, 128): each wave owns one 16x16 tile; a block's 8 waves
//    cover one row-tile across all 8 column tiles. h -> workspace.
//    A-frag: lane row = lane&15, kpair = (lane>>4)*2.  B[k][n] = W[n][k].
//    C/D: acc[i] holds row (rtile*16 + (lane>>4)*8 + i), col (lane&15).
// ---------------------------------------------------------------------------
__global__ __launch_bounds__(256) void mlp_wmma_kernel(const float* __restrict__ x,
                                                       const float* __restrict__ Wm,
                                                       const float* __restrict__ gamma,
                                                       const float* __restrict__ beta,
                                                       const float* __restrict__ mean,
                                                       const float* __restrict__ var,
                                                       float* __restrict__ h)
{
    const int wave  = threadIdx.x >> 5;
    const int lane  = threadIdx.x & 31;
    const int tile  = blockIdx.x * 8 + wave;   // 32768 tiles total
    const int rtile = tile >> 3;
    const int ctile = tile & 7;
    const int half  = lane >> 4;
    const int l16   = lane & 15;

    const int row = rtile * 16 + l16;          // A-fragment row for this lane
    const int col = ctile * 16 + l16;          // B-fragment / output column

    const float* xrow = x  + (size_t)row * CIN;
    const float* wrow = Wm + (size_t)col * CIN;

    v8f acc = {};
#pragma unroll
    for (int kk = 0; kk < CIN / 4; ++kk) {
        const int kb = kk * 4 + half * 2;
        v2f a = *(const v2f*)(xrow + kb);
        v2f bfrag = *(const v2f*)(wrow + kb);
        acc = __builtin_amdgcn_wmma_f32_16x16x4_f32(
            /*neg_a=*/false, a, /*neg_b=*/false, bfrag,
            /*c_mod=*/(short)0, acc, /*reuse_a=*/false, /*reuse_b=*/false);
    }

    // BN (eval) + ReLU: per-lane column constants
    const float g   = gamma[col];
    const float bt  = beta[col];
    const float mn  = mean[col];
    const float inv = __frsqrt_rn(var[col] + EPS_);

    float* hb = h + (size_t)(rtile * 16 + half * 8) * COUT + col;
#pragma unroll
    for (int i = 0; i < 8; ++i) {
        float v = ((acc[i] - mn) * inv) * g + bt;
        hb[(size_t)i * COUT] = fmaxf(v, 0.0f);
    }
}

// ---------------------------------------------------------------------------
// 3) KNN: one thread per query (b,m); point tiles staged through LDS with
//    precomputed |p|^2; register-resident top-16 (replace-max + rescan).
// ---------------------------------------------------------------------------
#define TILE_ 2048
__global__ __launch_bounds__(256) void knn_kernel(const float* __restrict__ p,
                                                  const float* __restrict__ p_out,
                                                  int* __restrict__ nbr)
{
    __shared__ float4 sp[TILE_];
    const int b = blockIdx.x >> 3;
    const int m = (blockIdx.x & 7) * 256 + threadIdx.x;
    const float* pb = p + (size_t)b * N_ * 3;
    const float* q  = p_out + ((size_t)b * M_ + m) * 3;
    const float qx = q[0], qy = q[1], qz = q[2];
    const float qn = __fadd_rn(__fadd_rn(__fmul_rn(qx, qx), __fmul_rn(qy, qy)),
                               __fmul_rn(qz, qz));

    float bd[KNN_]; int bi[KNN_];
#pragma unroll
    for (int k = 0; k < KNN_; ++k) { bd[k] = __builtin_inff(); bi[k] = 0; }
    float worst = __builtin_inff(); int wslot = 0;

    for (int t0 = 0; t0 < N_; t0 += TILE_) {
        __syncthreads();
        for (int s = 0; s < TILE_ / 256; ++s) {
            int li = threadIdx.x + s * 256;
            int gi = t0 + li;
            float xx = pb[gi * 3 + 0], yy = pb[gi * 3 + 1], zz = pb[gi * 3 + 2];
            float nn = __fadd_rn(__fadd_rn(__fmul_rn(xx, xx), __fmul_rn(yy, yy)),
                                 __fmul_rn(zz, zz));
            sp[li] = make_float4(xx, yy, zz, nn);
        }
        __syncthreads();
        for (int li = 0; li < TILE_; ++li) {
            float4 c = sp[li];
            float dot = __fadd_rn(__fadd_rn(__fmul_rn(qx, c.x), __fmul_rn(qy, c.y)),
                                  __fmul_rn(qz, c.z));
            float d2 = __fadd_rn(__fadd_rn(qn, c.w), __fmul_rn(-2.0f, dot));
            if (d2 < worst) {                 // strict < keeps earliest index on ties
                bd[wslot] = d2; bi[wslot] = t0 + li;
                worst = -__builtin_inff();
#pragma unroll
                for (int k = 0; k < KNN_; ++k)
                    if (bd[k] > worst) { worst = bd[k]; wslot = k; }
            }
        }
    }
    int* out = nbr + ((size_t)b * M_ + m) * KNN_;
#pragma unroll
    for (int k = 0; k < KNN_; ++k) out[k] = bi[k];
}

// ---------------------------------------------------------------------------
// 4) Gather neighbor features + max-pool over K. One block per (b,m).
// ---------------------------------------------------------------------------
__global__ __launch_bounds__(128) void pool_kernel(const float* __restrict__ h,
                                                   const int* __restrict__ nbr,
                                                   float* __restrict__ y)
{
    const int bm = blockIdx.x;         // b*M_ + m
    const int b  = bm / M_;
    const int c  = threadIdx.x;
    const int* nb = nbr + (size_t)bm * KNN_;
    float acc = -__builtin_inff();
#pragma unroll
    for (int k = 0; k < KNN_; ++k) {
        int r = nb[k];
        acc = fmaxf(acc, h[((size_t)b * N_ + r) * COUT + c]);
    }
    y[(size_t)bm * COUT + c] = acc;
}

// ---------------------------------------------------------------------------
extern "C" void kernel_launch(void* const* d_in, const int* in_sizes, int n_in,
                              void* d_out, int out_size, void* d_ws, size_t ws_size,
                              hipStream_t stream)
{
    const float* x     = (const float*)d_in[0];
    const float* p     = (const float*)d_in[1];
    const float* Wm    = (const float*)d_in[2];
    const float* gamma = (const float*)d_in[3];
    const float* beta  = (const float*)d_in[4];
    const float* rmean = (const float*)d_in[5];
    const float* rvar  = (const float*)d_in[6];

    float* y     = (float*)d_out;                          // (B,M,COUT)
    float* p_out = y + (size_t)B_ * M_ * COUT;             // (B,M,3)

    float* h   = (float*)d_ws;                             // (B*N, COUT) = 33.5 MB
    int*   nbr = (int*)((char*)d_ws + (size_t)B_ * N_ * COUT * sizeof(float)); // (B,M,K)

    const size_t fps_lds = (size_t)(4 * N_ + 2 * 1024 + 4) * sizeof(float);

    fps_kernel<<<B_, 1024, fps_lds, stream>>>(p, p_out);
    mlp_wmma_kernel<<<(B_ * N_ / 16), 256, 0, stream>>>(x, Wm, gamma, beta, rmean, rvar, h);
    knn_kernel<<<B_ * M_ / 256, 256, 0, stream>>>(p, p_out, nbr);
    pool_kernel<<<B_ * M_, 128, 0, stream>>>(h, nbr, y);
}